// GATModel_30339648979518
// MI455X (gfx1250) — compile-verified
//
#include <hip/hip_runtime.h>
#include <hip/hip_bf16.h>

// ---- problem constants (match reference) ----
#define GN   50000          // nodes
#define GE   800000         // edges
#define GEP  (GE + GN)      // edges + self loops
#define FIN  128
#define GH   8              // heads
#define GC   32             // channels/head
#define GHC  256            // H*C

typedef float v2f __attribute__((ext_vector_type(2)));
typedef float v8f __attribute__((ext_vector_type(8)));

// ---------------------------------------------------------------------------
// f32 WMMA GEMM:  Hout[N x 256] = X[N x K] * W[K x 256]
// One block = 256 threads = 8 waves; wave w computes col tiles w and w+8 of a
// 16-row strip, sharing its A fragment between the two tiles.
// ---------------------------------------------------------------------------
__global__ void gemm_f32_wmma(const float* __restrict__ X,
                              const float* __restrict__ W,
                              float* __restrict__ Hout, int K) {
  const int lane = threadIdx.x & 31;
  const int wv   = threadIdx.x >> 5;
  const int m0   = blockIdx.x * 16;
  const int row  = m0 + (lane & 15);        // A: lanes 0-15 / 16-31 both hold M=lane%16
  const int koff = (lane >> 4) * 2;         // A/B: upper half-wave holds K+2
  const int col  = lane & 15;
  const int n0a  = wv * 16;
  const int n0b  = (wv + 8) * 16;

  v8f acc0 = {};
  v8f acc1 = {};

  for (int k0 = 0; k0 < K; k0 += 4) {
    v2f a = *(const v2f*)(X + row * K + k0 + koff);   // K=koff, koff+1
    const float* wp = W + (size_t)(k0 + koff) * GHC;
    v2f b0, b1;
    b0.x = wp[n0a + col];         b0.y = wp[GHC + n0a + col];
    b1.x = wp[n0b + col];         b1.y = wp[GHC + n0b + col];
    acc0 = __builtin_amdgcn_wmma_f32_16x16x4_f32(false, a, false, b0,
                                                 (short)0, acc0, false, false);
    acc1 = __builtin_amdgcn_wmma_f32_16x16x4_f32(false, a, false, b1,
                                                 (short)0, acc1, false, false);
  }

  // D layout: VGPR j -> row m0 + j + 8*(lane>=16), col = n0 + lane%16
  const int rbase = m0 + (lane >> 4) * 8;
#pragma unroll
  for (int j = 0; j < 8; ++j) {
    Hout[(size_t)(rbase + j) * GHC + n0a + col] = acc0[j];
    Hout[(size_t)(rbase + j) * GHC + n0b + col] = acc1[j];
  }
}

// ---------------------------------------------------------------------------
// per-(node,head) attention logits: a_s = <h, att_src>, a_d = <h, att_dst>
// ---------------------------------------------------------------------------
__global__ void att_scores(const float* __restrict__ Hf,
                           const float* __restrict__ att_s,
                           const float* __restrict__ att_d,
                           float* __restrict__ a_s, float* __restrict__ a_d) {
  int idx = blockIdx.x * blockDim.x + threadIdx.x;
  if (idx >= GN * GH) return;
  int n = idx >> 3, head = idx & 7;
  const float* hp = Hf + (size_t)n * GHC + head * GC;
  const float* sp = att_s + head * GC;
  const float* dp = att_d + head * GC;
  float s = 0.f, d = 0.f;
#pragma unroll
  for (int c = 0; c < GC; ++c) {
    float v = hp[c];
    s += v * sp[c];
    d += v * dp[c];
  }
  a_s[idx] = s;
  a_d[idx] = d;
}

// init per-(node,head) max = -inf, denom = 0
__global__ void init_md(float* __restrict__ m, float* __restrict__ den) {
  int idx = blockIdx.x * blockDim.x + threadIdx.x;
  if (idx >= GN * GH) return;
  m[idx]   = __int_as_float(0xFF800000);   // -inf
  den[idx] = 0.f;
}

__global__ void zero_buf(float* __restrict__ p, int n) {
  int idx = blockIdx.x * blockDim.x + threadIdx.x;
  if (idx < n) p[idx] = 0.f;
}

// sign-split float atomic max -> native int/uint global atomics
__device__ __forceinline__ void atomicMaxFloat(float* addr, float val) {
  if (val >= 0.f)
    atomicMax((int*)addr, __float_as_int(val));
  else
    atomicMin((unsigned int*)addr, (unsigned int)__float_as_int(val));
}

__device__ __forceinline__ void edge_sd(const int* __restrict__ ei, int e,
                                        int& s, int& d) {
  if (e < GE) { s = ei[e]; d = ei[GE + e]; }
  else        { s = d = e - GE; }          // self loops
}

// pass 1: e = leaky_relu(a_s[src]+a_d[dst]); segment max over dst
__global__ void edge_pass1(const int* __restrict__ ei,
                           const float* __restrict__ a_s,
                           const float* __restrict__ a_d,
                           float* __restrict__ ebuf,
                           float* __restrict__ mbuf) {
  int idx = blockIdx.x * blockDim.x + threadIdx.x;
  if (idx >= GEP * GH) return;
  int e = idx >> 3, head = idx & 7;
  int s, d; edge_sd(ei, e, s, d);
  float v = a_s[s * GH + head] + a_d[d * GH + head];
  v = v > 0.f ? v : 0.2f * v;              // leaky relu, slope 0.2
  ebuf[idx] = v;
  atomicMaxFloat(mbuf + d * GH + head, v);
}

// pass 2: ex = exp(e - m[dst]); segment sum over dst
__global__ void edge_pass2(const int* __restrict__ ei,
                           float* __restrict__ ebuf,
                           const float* __restrict__ mbuf,
                           float* __restrict__ den) {
  int idx = blockIdx.x * blockDim.x + threadIdx.x;
  if (idx >= GEP * GH) return;
  int e = idx >> 3, head = idx & 7;
  int s, d; edge_sd(ei, e, s, d);
  float v = __expf(ebuf[idx] - mbuf[d * GH + head]);
  ebuf[idx] = v;
  atomicAdd(den + d * GH + head, v);
}

// pass 3: acc[dst] += alpha * h[src]   (one block per edge, one wave per head)
__global__ void edge_pass3(const int* __restrict__ ei,
                           const float* __restrict__ ebuf,
                           const float* __restrict__ den,
                           const float* __restrict__ Hf,
                           float* __restrict__ acc) {
  int e  = blockIdx.x;
  int ch = threadIdx.x;                    // 0..255, wave = one head
  int head = ch >> 5;
  int s, d; edge_sd(ei, e, s, d);
  float alpha = ebuf[e * GH + head] / (den[d * GH + head] + 1e-16f);
  atomicAdd(acc + (size_t)d * GHC + ch, alpha * Hf[(size_t)s * GHC + ch]);
}

// out = relu(acc + bias)
__global__ void relu_bias(const float* __restrict__ acc,
                          const float* __restrict__ bias,
                          float* __restrict__ out) {
  int idx = blockIdx.x * blockDim.x + threadIdx.x;
  if (idx >= GN * GHC) return;
  float v = acc[idx] + bias[idx & (GHC - 1)];
  out[idx] = v > 0.f ? v : 0.f;
}

// edge head: out[e] = <h[src]*h[dst], We> + be   (one wave per edge)
__global__ void edge_pred_k(const int* __restrict__ ei,
                            const float* __restrict__ Hf,
                            const float* __restrict__ We,
                            const float* __restrict__ be,
                            float* __restrict__ out) {
  int lane = threadIdx.x & 31, wv = threadIdx.x >> 5;
  int e = blockIdx.x * 8 + wv;
  if (e >= GE) return;
  int s = ei[e], d = ei[GE + e];
  float sum = 0.f;
#pragma unroll
  for (int j = 0; j < 8; ++j) {
    int c = lane + 32 * j;
    sum += Hf[(size_t)s * GHC + c] * Hf[(size_t)d * GHC + c] * We[c];
  }
#pragma unroll
  for (int off = 16; off > 0; off >>= 1) sum += __shfl_xor(sum, off, 32);
  if (lane == 0) out[e] = sum + be[0];
}

// node head: out[E+n] = <h[n], Wn> + bn   (one wave per node)
__global__ void node_pred_k(const float* __restrict__ Hf,
                            const float* __restrict__ Wn,
                            const float* __restrict__ bn,
                            float* __restrict__ out) {
  int lane = threadIdx.x & 31, wv = threadIdx.x >> 5;
  int n = blockIdx.x * 8 + wv;
  if (n >= GN) return;
  float sum = 0.f;
#pragma unroll
  for (int j = 0; j < 8; ++j) {
    int c = lane + 32 * j;
    sum += Hf[(size_t)n * GHC + c] * Wn[c];
  }
#pragma unroll
  for (int off = 16; off > 0; off >>= 1) sum += __shfl_xor(sum, off, 32);
  if (lane == 0) out[GE + n] = sum + bn[0];
}

// ---------------------------------------------------------------------------
extern "C" void kernel_launch(void* const* d_in, const int* in_sizes, int n_in,
                              void* d_out, int out_size, void* d_ws, size_t ws_size,
                              hipStream_t stream) {
  const float* x        = (const float*)d_in[0];
  const int*   ei       = (const int*)d_in[1];
  // d_in[2] = edge_weights (unused by reference)
  const float* W1       = (const float*)d_in[3];
  const float* att_src1 = (const float*)d_in[4];
  const float* att_dst1 = (const float*)d_in[5];
  const float* b1       = (const float*)d_in[6];
  const float* W2       = (const float*)d_in[7];
  const float* att_src2 = (const float*)d_in[8];
  const float* att_dst2 = (const float*)d_in[9];
  const float* b2       = (const float*)d_in[10];
  const float* We       = (const float*)d_in[11];
  const float* be       = (const float*)d_in[12];
  const float* Wn       = (const float*)d_in[13];
  const float* bn       = (const float*)d_in[14];
  float* out = (float*)d_out;

  // workspace layout (floats)
  float* ws   = (float*)d_ws;
  float* bufA = ws;                                   // N*256
  float* bufB = bufA + (size_t)GN * GHC;              // N*256
  float* a_s  = bufB + (size_t)GN * GHC;              // N*8
  float* a_d  = a_s + (size_t)GN * GH;                // N*8
  float* mbuf = a_d + (size_t)GN * GH;                // N*8
  float* den  = mbuf + (size_t)GN * GH;               // N*8
  float* ebuf = den + (size_t)GN * GH;                // (E+N)*8

  const int nhBlocks  = (GN * GH + 255) / 256;        // per-(node,head) grids
  const int ehBlocks  = (GEP * GH + 255) / 256;       // per-(edge,head) grids
  const int nc        = GN * GHC;
  const int ncBlocks  = (nc + 255) / 256;
  const int rowTiles  = GN / 16;                      // 3125, exact

  // ---------------- layer 1 ----------------
  gemm_f32_wmma<<<rowTiles, 256, 0, stream>>>(x, W1, bufA, FIN);
  att_scores<<<nhBlocks, 256, 0, stream>>>(bufA, att_src1, att_dst1, a_s, a_d);
  init_md<<<nhBlocks, 256, 0, stream>>>(mbuf, den);
  zero_buf<<<ncBlocks, 256, 0, stream>>>(bufB, nc);
  edge_pass1<<<ehBlocks, 256, 0, stream>>>(ei, a_s, a_d, ebuf, mbuf);
  edge_pass2<<<ehBlocks, 256, 0, stream>>>(ei, ebuf, mbuf, den);
  edge_pass3<<<GEP, 256, 0, stream>>>(ei, ebuf, den, bufA, bufB);
  relu_bias<<<ncBlocks, 256, 0, stream>>>(bufB, b1, bufA);      // h1 -> bufA

  // ---------------- layer 2 ----------------
  gemm_f32_wmma<<<rowTiles, 256, 0, stream>>>(bufA, W2, bufB, GHC);
  att_scores<<<nhBlocks, 256, 0, stream>>>(bufB, att_src2, att_dst2, a_s, a_d);
  init_md<<<nhBlocks, 256, 0, stream>>>(mbuf, den);
  zero_buf<<<ncBlocks, 256, 0, stream>>>(bufA, nc);
  edge_pass1<<<ehBlocks, 256, 0, stream>>>(ei, a_s, a_d, ebuf, mbuf);
  edge_pass2<<<ehBlocks, 256, 0, stream>>>(ei, ebuf, mbuf, den);
  edge_pass3<<<GEP, 256, 0, stream>>>(ei, ebuf, den, bufB, bufA);
  relu_bias<<<ncBlocks, 256, 0, stream>>>(bufA, b2, bufB);      // h2 -> bufB

  // ---------------- heads ----------------
  edge_pred_k<<<GE / 8, 256, 0, stream>>>(ei, bufB, We, be, out);
  node_pred_k<<<GN / 8, 256, 0, stream>>>(bufB, Wn, bn, out);   // writes out+GE
}